// NodeNetwork_26182120636656
// MI455X (gfx1250) — compile-verified
//
#include <hip/hip_runtime.h>

typedef float v2f __attribute__((ext_vector_type(2)));
typedef float v8f __attribute__((ext_vector_type(8)));

#define DD 32   // node feature dim
#define OO 64   // hidden/output dim
#define MSTRIDE 100   // 16x96 M tile, padded (even -> b64 aligned, 100%64=36 decorrelates banks)
#define W1STRIDE 98   // 64x96 W1^T, padded (even, 98%64=34)
#define W2STRIDE 66   // 64x64 W2^T, padded (even, 66%64=2)
#define HSTRIDE  68   // 16x64 h tile, padded (even, 68%64=4)

// ---------------------------------------------------------------- zero ws ---
__global__ void zero_ws_kernel(float* __restrict__ p, long n) {
    long i = (long)blockIdx.x * blockDim.x + threadIdx.x;
    long stride = (long)gridDim.x * blockDim.x;
    for (; i < n; i += stride) p[i] = 0.0f;
}

// ------------------------------------------------- edge gather/scatter-add ---
// one wave (32 lanes) per edge; lane d handles feature d.
// mi[col] += w * x[row];  mo[row] += w * x[col]
__global__ void edge_scatter_kernel(const float* __restrict__ x,
                                    const long long* __restrict__ ei,
                                    const float* __restrict__ ea,
                                    float* __restrict__ mi,
                                    float* __restrict__ mo,
                                    long E) {
    long gid = (long)blockIdx.x * blockDim.x + threadIdx.x;
    long e = gid >> 5;
    int  d = (int)(gid & 31);
    if (e >= E) return;
    int row = (int)ei[e];       // edge_index[0][e]
    int col = (int)ei[E + e];   // edge_index[1][e]
    float w  = ea[e];
    float xr = x[(long)row * DD + d];
    float xc = x[(long)col * DD + d];
    atomicAdd(&mi[(long)col * DD + d], w * xr);
    atomicAdd(&mo[(long)row * DD + d], w * xc);
}

// ------------------------------------------------------------ fused MLP -----
// one wave per 16-node tile; 2 waves per 64-thread block.
// h = tanh([mi|mo|x] @ W1 + b1); out = tanh(h @ W2 + b2)
// all GEMMs via V_WMMA_F32_16X16X4_F32 (f32 in, f32 accumulate).
__global__ __launch_bounds__(64) void mlp_wmma_kernel(
    const float* __restrict__ x,
    const float* __restrict__ mi,
    const float* __restrict__ mo,
    const float* __restrict__ W1,   // [96][64] row-major (k-major)
    const float* __restrict__ b1,   // [64]
    const float* __restrict__ W2,   // [64][64] row-major
    const float* __restrict__ b2,   // [64]
    float* __restrict__ out,        // [N][64]
    int N, int ntiles)
{
    __shared__ float sM[2][16][MSTRIDE];    // per-wave [mi|mo|x] tile
    __shared__ float sH[2][16][HSTRIDE];    // per-wave hidden tile
    __shared__ float sW1[OO][W1STRIDE];     // W1^T : [o][k]
    __shared__ float sW2[OO][W2STRIDE];     // W2^T : [o][k]
    __shared__ float sb1[OO];
    __shared__ float sb2[OO];

    const int tid  = threadIdx.x;
    const int wv   = tid >> 5;
    const int lane = tid & 31;
    const int half = lane >> 4;     // 0: K=0,1 ; 1: K=2,3 (per 16x16x4 layout)
    const int l15  = lane & 15;
    const int tile = blockIdx.x * 2 + wv;
    const int node0 = tile * 16;

    // ---- stage W1^T, W2^T, biases (whole block, coalesced global reads) ----
    for (int i = tid; i < 96 * OO; i += 64) sW1[i & 63][i >> 6] = W1[i];
    for (int i = tid; i < OO * OO; i += 64) sW2[i & 63][i >> 6] = W2[i];
    sb1[tid] = b1[tid];
    sb2[tid] = b2[tid];

    // ---- stage M = [mi | mo | x] tile (16 rows x 96 cols) ----
#pragma unroll 4
    for (int r = 0; r < 16; ++r) {
        int node = node0 + r; if (node > N - 1) node = N - 1;
        long base = (long)node * DD;
        sM[wv][r][lane]          = mi[base + lane];
        sM[wv][r][DD + lane]     = mo[base + lane];
        sM[wv][r][2 * DD + lane] = x[base + lane];
    }
    __syncthreads();

    // ---- GEMM1: h_acc[16x64] = M[16x96] @ W1[96x64] ----
    v8f a0 = {}, a1 = {}, a2 = {}, a3 = {};
    for (int k = 0; k < 96; k += 4) {
        int kk = k + 2 * half;
        v2f av  = *(const v2f*)&sM[wv][l15][kk];     // A: M[l15][kk..kk+1]
        v2f bv0 = *(const v2f*)&sW1[l15][kk];        // B: W1[kk..kk+1][col]
        v2f bv1 = *(const v2f*)&sW1[16 + l15][kk];
        v2f bv2 = *(const v2f*)&sW1[32 + l15][kk];
        v2f bv3 = *(const v2f*)&sW1[48 + l15][kk];
        a0 = __builtin_amdgcn_wmma_f32_16x16x4_f32(false, av, false, bv0, (short)0, a0, false, false);
        a1 = __builtin_amdgcn_wmma_f32_16x16x4_f32(false, av, false, bv1, (short)0, a1, false, false);
        a2 = __builtin_amdgcn_wmma_f32_16x16x4_f32(false, av, false, bv2, (short)0, a2, false, false);
        a3 = __builtin_amdgcn_wmma_f32_16x16x4_f32(false, av, false, bv3, (short)0, a3, false, false);
    }

    // ---- bias + tanh, stage h into LDS (C layout -> row-major) ----
    {
        float c0 = sb1[l15], c1 = sb1[16 + l15], c2 = sb1[32 + l15], c3 = sb1[48 + l15];
#pragma unroll
        for (int r = 0; r < 8; ++r) {
            int rr = r + 8 * half;                    // C layout: rows r / r+8 per half
            sH[wv][rr][l15]      = tanhf(a0[r] + c0);
            sH[wv][rr][16 + l15] = tanhf(a1[r] + c1);
            sH[wv][rr][32 + l15] = tanhf(a2[r] + c2);
            sH[wv][rr][48 + l15] = tanhf(a3[r] + c3);
        }
    }
    __syncthreads();

    // ---- GEMM2: out_acc[16x64] = h[16x64] @ W2[64x64] ----
    v8f z = {};
    a0 = z; a1 = z; a2 = z; a3 = z;
    for (int k = 0; k < 64; k += 4) {
        int kk = k + 2 * half;
        v2f av  = *(const v2f*)&sH[wv][l15][kk];
        v2f bv0 = *(const v2f*)&sW2[l15][kk];
        v2f bv1 = *(const v2f*)&sW2[16 + l15][kk];
        v2f bv2 = *(const v2f*)&sW2[32 + l15][kk];
        v2f bv3 = *(const v2f*)&sW2[48 + l15][kk];
        a0 = __builtin_amdgcn_wmma_f32_16x16x4_f32(false, av, false, bv0, (short)0, a0, false, false);
        a1 = __builtin_amdgcn_wmma_f32_16x16x4_f32(false, av, false, bv1, (short)0, a1, false, false);
        a2 = __builtin_amdgcn_wmma_f32_16x16x4_f32(false, av, false, bv2, (short)0, a2, false, false);
        a3 = __builtin_amdgcn_wmma_f32_16x16x4_f32(false, av, false, bv3, (short)0, a3, false, false);
    }

    // ---- bias + tanh, store output (guards only AFTER the last WMMA) ----
    if (tile < ntiles) {
        float d0 = sb2[l15], d1 = sb2[16 + l15], d2 = sb2[32 + l15], d3 = sb2[48 + l15];
#pragma unroll
        for (int r = 0; r < 8; ++r) {
            int node = node0 + r + 8 * half;
            if (node < N) {
                float* op = out + (long)node * OO;
                op[l15]      = tanhf(a0[r] + d0);
                op[16 + l15] = tanhf(a1[r] + d1);
                op[32 + l15] = tanhf(a2[r] + d2);
                op[48 + l15] = tanhf(a3[r] + d3);
            }
        }
    }
}

// ---------------------------------------------------------------- launch ----
extern "C" void kernel_launch(void* const* d_in, const int* in_sizes, int n_in,
                              void* d_out, int out_size, void* d_ws, size_t ws_size,
                              hipStream_t stream) {
    const float*     x  = (const float*)d_in[0];
    const long long* ei = (const long long*)d_in[1];
    const float*     ea = (const float*)d_in[2];
    const float*     W1 = (const float*)d_in[3];
    const float*     b1 = (const float*)d_in[4];
    const float*     W2 = (const float*)d_in[5];
    const float*     b2 = (const float*)d_in[6];
    float* out = (float*)d_out;

    const long N = in_sizes[0] / DD;
    const long E = in_sizes[2];

    float* mi = (float*)d_ws;          // [N][32]
    float* mo = mi + N * DD;           // [N][32]

    // 1) zero the scatter accumulators
    zero_ws_kernel<<<1024, 256, 0, stream>>>(mi, 2 * N * DD);

    // 2) edge gather/scatter: one wave per edge
    long ethreads = E * 32;
    int  eblocks  = (int)((ethreads + 255) / 256);
    edge_scatter_kernel<<<eblocks, 256, 0, stream>>>(x, ei, ea, mi, mo, E);

    // 3) fused 2-layer MLP with f32 WMMA
    int ntiles  = (int)((N + 15) / 16);
    int mblocks = (ntiles + 1) / 2;
    mlp_wmma_kernel<<<mblocks, 64, 0, stream>>>(x, mi, mo, W1, b1, W2, b2, out, (int)N, ntiles);
}